// DeepFM_79989470920946
// MI455X (gfx1250) — compile-verified
//
#include <hip/hip_runtime.h>

// Problem constants (match reference)
#define BATCH   4096
#define NFIELD  16
#define DDIM    256
#define H1DIM   1024
#define H2DIM   512
#define HODIM   64
#define PAIRS   120
#define NCLS    5

typedef __attribute__((ext_vector_type(2))) float v2f;
typedef __attribute__((ext_vector_type(8))) float v8f;

static __device__ __forceinline__ v8f wmma_f32(v2f a, v2f b, v8f c) {
    // D = A(16x4) * B(4x16) + C(16x16), full fp32
    return __builtin_amdgcn_wmma_f32_16x16x4_f32(
        /*neg_a=*/false, a, /*neg_b=*/false, b,
        /*c_mod=*/(short)0, c, /*reuse_a=*/false, /*reuse_b=*/false);
}

// ---------------------------------------------------------------------------
// Kernel 1: per-batch Gram matrix via WMMA, extract upper triangle (k=1).
// One wave (32 lanes) per batch; 8 waves per block.
// A-tile layout (32-bit 16x4): lane = M + 16*(K>=2), vgpr = K%2.
// B-tile layout (4x16): lane = N + 16*(K>=2)?? -> symmetric striping; for
// X*X^T the per-lane contents of A and B tiles coincide, so one float2 load
// feeds both operands.
// ---------------------------------------------------------------------------
__global__ __launch_bounds__(256) void gram_pairs_kernel(
    const float* __restrict__ x,      // [BATCH, NFIELD, DDIM]
    float* __restrict__ second)       // [BATCH, PAIRS]
{
    __shared__ float lds[8][NFIELD][NFIELD + 1];

    const int wave = threadIdx.x >> 5;
    const int lane = threadIdx.x & 31;
    const int b    = blockIdx.x * 8 + wave;

    const int row   = lane & 15;          // M (== N) index
    const int khalf = (lane >> 4) << 1;   // 0 for lanes 0-15, 2 for lanes 16-31

    const float* xp = x + (size_t)b * NFIELD * DDIM + (size_t)row * DDIM + khalf;

    v8f acc = {};
#pragma unroll 8
    for (int k = 0; k < DDIM; k += 4) {
        v2f av = *(const v2f*)(xp + k);   // X[row, k+khalf], X[row, k+khalf+1]
        acc = wmma_f32(av, av, acc);      // A and B registers identical for X*X^T
    }

    // C/D layout: vgpr r -> M = r + 8*(lane>=16); N = lane&15
    const int mbase = (lane >> 4) * 8;
#pragma unroll
    for (int r = 0; r < 8; ++r) {
        lds[wave][mbase + r][row] = acc[r];
    }
    __syncthreads();

    // triu_indices(16, k=1) row-major: (0,1)..(0,15),(1,2)..(14,15)
    for (int p = lane; p < PAIRS; p += 32) {
        int i = 0, rem = p;
        while (rem >= (NFIELD - 1 - i)) { rem -= (NFIELD - 1 - i); ++i; }
        int j = i + 1 + rem;
        second[(size_t)b * PAIRS + p] = lds[wave][i][j];
    }
}

// ---------------------------------------------------------------------------
// Kernel 2: fp32 GEMM  C = act(A @ W + bias)
//   A: [M, K] row-major, W: [K, N] row-major, C: [M, N] row-major.
// Each wave computes a 64x16 tile of C (4 accumulators); B operand is reused
// across the 4 WMMAs per K-step. All dims multiples of 64/16/4.
// ---------------------------------------------------------------------------
__global__ __launch_bounds__(256) void gemm_bias_act_kernel(
    const float* __restrict__ A, const float* __restrict__ W,
    const float* __restrict__ bias, float* __restrict__ C,
    int M, int K, int Nt, int do_relu)
{
    const int gwave = (int)((blockIdx.x * blockDim.x + threadIdx.x) >> 5);
    const int lane  = threadIdx.x & 31;

    const int ntn = Nt >> 4;            // # of 16-wide column tiles
    const int tm  = gwave / ntn;        // 64-row tile index
    const int tn  = gwave % ntn;        // 16-col tile index

    const int rowN  = lane & 15;
    const int khalf = (lane >> 4) << 1; // 0 or 2
    const int n     = tn * 16 + rowN;

    const float* a0 = A + (size_t)(tm * 64 +  0 + rowN) * K + khalf;
    const float* a1 = A + (size_t)(tm * 64 + 16 + rowN) * K + khalf;
    const float* a2 = A + (size_t)(tm * 64 + 32 + rowN) * K + khalf;
    const float* a3 = A + (size_t)(tm * 64 + 48 + rowN) * K + khalf;
    const float* wp = W + (size_t)khalf * Nt + n;

    v8f acc0 = {}, acc1 = {}, acc2 = {}, acc3 = {};

#pragma unroll 4
    for (int k = 0; k < K; k += 4) {
        // B tile: vgpr0 = W[k+khalf, n], vgpr1 = W[k+khalf+1, n]
        v2f bv;
        bv.x = wp[(size_t)k * Nt];
        bv.y = wp[(size_t)(k + 1) * Nt];
        v2f av0 = *(const v2f*)(a0 + k);
        v2f av1 = *(const v2f*)(a1 + k);
        v2f av2 = *(const v2f*)(a2 + k);
        v2f av3 = *(const v2f*)(a3 + k);
        acc0 = wmma_f32(av0, bv, acc0);
        acc1 = wmma_f32(av1, bv, acc1);
        acc2 = wmma_f32(av2, bv, acc2);
        acc3 = wmma_f32(av3, bv, acc3);
    }

    const float bn = bias[n];
    float* cp = C + (size_t)(tm * 64) * Nt + n;
    const int mb = (lane >> 4) * 8;     // C layout: vgpr r -> M = r + 8*(lane>=16)
#pragma unroll
    for (int r = 0; r < 8; ++r) {
        float v0 = acc0[r] + bn;
        float v1 = acc1[r] + bn;
        float v2 = acc2[r] + bn;
        float v3 = acc3[r] + bn;
        if (do_relu) {
            v0 = fmaxf(v0, 0.0f); v1 = fmaxf(v1, 0.0f);
            v2 = fmaxf(v2, 0.0f); v3 = fmaxf(v3, 0.0f);
        }
        cp[(size_t)(mb + r +  0) * Nt] = v0;
        cp[(size_t)(mb + r + 16) * Nt] = v1;
        cp[(size_t)(mb + r + 32) * Nt] = v2;
        cp[(size_t)(mb + r + 48) * Nt] = v3;
    }
}

// ---------------------------------------------------------------------------
// Kernel 3: final projection  out = [higher | second] @ Wc + bc   (184 -> 5)
// Wc/bc staged in LDS; one thread per batch row.
// ---------------------------------------------------------------------------
__global__ __launch_bounds__(256) void final_proj_kernel(
    const float* __restrict__ higher,  // [BATCH, HODIM]
    const float* __restrict__ second,  // [BATCH, PAIRS]
    const float* __restrict__ Wc,      // [HODIM+PAIRS, NCLS]
    const float* __restrict__ bc,      // [NCLS]
    float* __restrict__ out)           // [BATCH, NCLS]
{
    __shared__ float wc[(HODIM + PAIRS) * NCLS];
    __shared__ float bcs[NCLS];

    for (int i = threadIdx.x; i < (HODIM + PAIRS) * NCLS; i += blockDim.x)
        wc[i] = Wc[i];
    if (threadIdx.x < NCLS) bcs[threadIdx.x] = bc[threadIdx.x];
    __syncthreads();

    const int b = blockIdx.x * blockDim.x + threadIdx.x;

    float acc[NCLS];
#pragma unroll
    for (int c = 0; c < NCLS; ++c) acc[c] = bcs[c];

    const float* hp = higher + (size_t)b * HODIM;
#pragma unroll 4
    for (int j = 0; j < HODIM; ++j) {
        float h = hp[j];
#pragma unroll
        for (int c = 0; c < NCLS; ++c) acc[c] = fmaf(h, wc[j * NCLS + c], acc[c]);
    }
    const float* sp = second + (size_t)b * PAIRS;
#pragma unroll 4
    for (int j = 0; j < PAIRS; ++j) {
        float s = sp[j];
#pragma unroll
        for (int c = 0; c < NCLS; ++c) acc[c] = fmaf(s, wc[(HODIM + j) * NCLS + c], acc[c]);
    }
#pragma unroll
    for (int c = 0; c < NCLS; ++c) out[(size_t)b * NCLS + c] = acc[c];
}

// ---------------------------------------------------------------------------
extern "C" void kernel_launch(void* const* d_in, const int* in_sizes, int n_in,
                              void* d_out, int out_size, void* d_ws, size_t ws_size,
                              hipStream_t stream) {
    const float* x  = (const float*)d_in[0];
    const float* W1 = (const float*)d_in[1];
    const float* b1 = (const float*)d_in[2];
    const float* W2 = (const float*)d_in[3];
    const float* b2 = (const float*)d_in[4];
    const float* W3 = (const float*)d_in[5];
    const float* b3 = (const float*)d_in[6];
    const float* Wc = (const float*)d_in[7];
    const float* bc = (const float*)d_in[8];
    float* out = (float*)d_out;

    // Workspace carve-up (all 256B-aligned sizes)
    char* ws = (char*)d_ws;
    float* second = (float*)(ws);                                  // 4096*120*4   = 1,966,080
    float* h1     = (float*)(ws + 1966080);                        // 4096*1024*4  = 16,777,216
    float* h2     = (float*)(ws + 1966080 + 16777216);             // 4096*512*4   = 8,388,608
    float* higher = (float*)(ws + 1966080 + 16777216 + 8388608);   // 4096*64*4    = 1,048,576

    // 1) Gram + pair extraction: one wave per batch, 8 waves/block
    gram_pairs_kernel<<<BATCH / 8, 256, 0, stream>>>(x, second);

    // 2) h1 = relu(flat @ W1 + b1): waves = (4096/64)*(1024/16) = 4096 -> 512 blocks
    gemm_bias_act_kernel<<<512, 256, 0, stream>>>(x, W1, b1, h1,
                                                  BATCH, NFIELD * DDIM, H1DIM, 1);

    // 3) h2 = relu(h1 @ W2 + b2): waves = 64*32 = 2048 -> 256 blocks
    gemm_bias_act_kernel<<<256, 256, 0, stream>>>(h1, W2, b2, h2,
                                                  BATCH, H1DIM, H2DIM, 1);

    // 4) higher = h2 @ W3 + b3: waves = 64*4 = 256 -> 32 blocks
    gemm_bias_act_kernel<<<32, 256, 0, stream>>>(h2, W3, b3, higher,
                                                 BATCH, H2DIM, HODIM, 0);

    // 5) out = [higher | second] @ Wc + bc
    final_proj_kernel<<<BATCH / 256, 256, 0, stream>>>(higher, second, Wc, bc, out);
}